// GCN_13735305413410
// MI455X (gfx1250) — compile-verified
//
#include <hip/hip_runtime.h>
#include <hip/hip_bf16.h>

typedef float v2f __attribute__((ext_vector_type(2)));
typedef float v8f __attribute__((ext_vector_type(8)));

#define N_NODES 100000
#define N_EDGES 1600000
#define KDIM    128
#define HIDC    128
#define OUTC    40
#define BN_EPS  1e-5f

// ---------------------------------------------------------------- utilities
__global__ void fill_kernel(float* __restrict__ p, float val, int n) {
    int i = blockIdx.x * blockDim.x + threadIdx.x;
    if (i < n) p[i] = val;
}

__global__ void degree_kernel(const int* __restrict__ dst, float* __restrict__ deg, int e) {
    int i = blockIdx.x * blockDim.x + threadIdx.x;
    if (i < e) atomicAdd(&deg[dst[i]], 1.0f);
}

__global__ void rsqrt_inplace_kernel(float* __restrict__ d, int n) {
    int i = blockIdx.x * blockDim.x + threadIdx.x;
    if (i < n) d[i] = rsqrtf(d[i]);   // deg >= 1 always (self loop)
}

// ---------------------------------------------------------------- W packing
// Wp[(s*CPAD + n)*4 + j] = W[(4s+j)*COUT + n]  (zero-padded n >= COUT)
// So a lane's B-fragment for k-step s is a single b64 load:
//   b = *(v2f*)(Wp + (n*4 + 2*half) + s*CPAD*4)
template<int COUT, int CPAD>
__global__ void pack_w_kernel(const float* __restrict__ W, float* __restrict__ Wp) {
    constexpr int TOT = (KDIM / 4) * CPAD * 4;
    int i = blockIdx.x * blockDim.x + threadIdx.x;
    if (i >= TOT) return;
    int j = i & 3;
    int n = (i >> 2) % CPAD;
    int s = (i >> 2) / CPAD;
    int k = 4 * s + j;
    Wp[i] = (n < COUT) ? W[k * COUT + n] : 0.0f;
}

// ---------------------------------------------------------------- WMMA GEMM
// H[m,n] = dinv[m] * sum_k A[m,k]*W[k,n].  One wave computes a 16 x (16*NT)
// strip with V_WMMA_F32_16X16X4_F32; K fully unrolled; all loads are
// base + compile-time-immediate b64 loads; 32-bit address math only.
template<int COUT, int CPAD, int NT>
__global__ __launch_bounds__(256)
void gemm_wmma_f32(const float* __restrict__ A, const float* __restrict__ Wp,
                   const float* __restrict__ dinv, float* __restrict__ H, int M)
{
    constexpr int NGRP = CPAD / (16 * NT);     // N-strip groups
    int wave  = (blockIdx.x * blockDim.x + threadIdx.x) >> 5;
    int lane  = threadIdx.x & 31;
    int tileM = wave / NGRP;
    int grpN  = wave - tileM * NGRP;
    if (tileM * 16 >= M) return;               // wave-uniform: EXEC stays all-ones

    const int half = lane >> 4;                // 0/1
    const int l15  = lane & 15;
    const int row  = tileM * 16 + l15;

    const float* __restrict__ arow  = A  + row * KDIM;
    const float* __restrict__ bbase = Wp + (grpN * NT * 16 + l15) * 4 + 2 * half;

    v8f acc[NT];
#pragma unroll
    for (int t = 0; t < NT; ++t) acc[t] = (v8f){0.f,0.f,0.f,0.f,0.f,0.f,0.f,0.f};

#pragma unroll
    for (int s = 0; s < KDIM / 4; ++s) {
        v2f a = *(const v2f*)(arow + 4 * s + 2 * half);
#pragma unroll
        for (int t = 0; t < NT; ++t) {
            v2f b = *(const v2f*)(bbase + s * CPAD * 4 + t * 64);
            acc[t] = __builtin_amdgcn_wmma_f32_16x16x4_f32(
                         false, a, false, b, (short)0, acc[t], false, false);
        }
    }

#pragma unroll
    for (int t = 0; t < NT; ++t) {
        int col = grpN * NT * 16 + t * 16 + l15;
#pragma unroll
        for (int r = 0; r < 8; ++r) {
            int m = tileM * 16 + r + 8 * half;
            float v = acc[t][r] * dinv[m];
            if (COUT == CPAD || col < COUT)
                H[m * COUT + col] = v;
        }
    }
}

// ---------------------------------------------------------------- scatter
// acc[dst] += h_scaled[src]; one float4 chunk per thread; C4 chunks per edge.
template<int C4>
__global__ __launch_bounds__(256)
void scatter_kernel(const float4* __restrict__ hs, const int* __restrict__ src,
                    const int* __restrict__ dst, float* __restrict__ acc, int e)
{
    int tid = blockIdx.x * blockDim.x + threadIdx.x;
    if (tid >= e * C4) return;                 // max 51.2M, fits int
    int ed = tid / C4;
    int c  = tid - ed * C4;
    int s  = src[ed];
    int d  = dst[ed];
    float4 v = hs[s * C4 + c];
    float* p = acc + (d * C4 + c) * 4;
    atomicAdd(p + 0, v.x);
    atomicAdd(p + 1, v.y);
    atomicAdd(p + 2, v.z);
    atomicAdd(p + 3, v.w);
}

// ---------------------------------------------------------------- finalize
// v = dinv[i]*(acc[i,c] + hs[i,c]) + bias[c]; write back; accumulate BN sums.
__global__ __launch_bounds__(256)
void finalize_bn_stats_kernel(const float* __restrict__ hs, float* __restrict__ acc,
                              const float* __restrict__ dinv, const float* __restrict__ bias,
                              float* __restrict__ stats, int n)
{
    __shared__ float s_sum[HIDC];
    __shared__ float s_sq[HIDC];
    for (int i = threadIdx.x; i < HIDC; i += blockDim.x) { s_sum[i] = 0.f; s_sq[i] = 0.f; }
    __syncthreads();

    int total  = n * HIDC;                     // 12.8M
    int stride = gridDim.x * blockDim.x;       // multiple of 128
    for (int idx = blockIdx.x * blockDim.x + threadIdx.x; idx < total; idx += stride) {
        int i = idx >> 7;
        int c = idx & (HIDC - 1);
        float v = dinv[i] * (acc[idx] + hs[idx]) + bias[c];
        acc[idx] = v;
        atomicAdd(&s_sum[c], v);
        atomicAdd(&s_sq[c], v * v);
    }
    __syncthreads();
    for (int i = threadIdx.x; i < HIDC; i += blockDim.x) {
        atomicAdd(&stats[i],        s_sum[i]);
        atomicAdd(&stats[HIDC + i], s_sq[i]);
    }
}

// layer-3 finalize: bias only, no BN stats
__global__ __launch_bounds__(256)
void finalize_out_kernel(const float* __restrict__ hs, float* __restrict__ out,
                         const float* __restrict__ dinv, const float* __restrict__ bias,
                         int n)
{
    int idx = blockIdx.x * blockDim.x + threadIdx.x;
    if (idx >= n * OUTC) return;
    int i = idx / OUTC;
    int c = idx - i * OUTC;
    out[idx] = dinv[i] * (out[idx] + hs[idx]) + bias[c];
}

// ---------------------------------------------------------------- BN + ReLU
__global__ __launch_bounds__(256)
void bn_relu_kernel(float* __restrict__ x, const float* __restrict__ stats,
                    const float* __restrict__ gamma, const float* __restrict__ beta,
                    int n)
{
    int idx = blockIdx.x * blockDim.x + threadIdx.x;
    if (idx >= n * HIDC) return;
    int c = idx & (HIDC - 1);
    float invn = 1.0f / (float)n;
    float mean = stats[c] * invn;
    float var  = stats[HIDC + c] * invn - mean * mean;
    float v = (x[idx] - mean) * rsqrtf(var + BN_EPS) * gamma[c] + beta[c];
    x[idx] = v > 0.f ? v : 0.f;
}

// ---------------------------------------------------------------- launch
static inline int cdiv(long long a, long long b) { return (int)((a + b - 1) / b); }

extern "C" void kernel_launch(void* const* d_in, const int* in_sizes, int n_in,
                              void* d_out, int out_size, void* d_ws, size_t ws_size,
                              hipStream_t stream)
{
    const float* x   = (const float*)d_in[0];
    const int*   ei  = (const int*)d_in[1];
    const float* W1  = (const float*)d_in[2];
    const float* b1  = (const float*)d_in[3];
    const float* g1  = (const float*)d_in[4];
    const float* bt1 = (const float*)d_in[5];
    const float* W2  = (const float*)d_in[6];
    const float* b2  = (const float*)d_in[7];
    const float* g2  = (const float*)d_in[8];
    const float* bt2 = (const float*)d_in[9];
    const float* W3  = (const float*)d_in[10];
    const float* b3  = (const float*)d_in[11];
    float* out = (float*)d_out;

    const int* src = ei;
    const int* dst = ei + N_EDGES;

    char* ws = (char*)d_ws;
    float* dinv  = (float*)(ws);                         // 100000 f32
    float* stats = (float*)(ws + (512ll << 10));         // 256 f32
    float* Wp    = (float*)(ws + (576ll << 10));         // packed W, <= 64 KB
    float* bufA  = (float*)(ws + (1ll   << 20));         // 51.2 MB (h_scaled)
    float* bufB  = (float*)(ws + (54ll  << 20));         // 51.2 MB (acc / features)

    const int NC = N_NODES * HIDC;                       // 12.8M
    const int NO = N_NODES * OUTC;                       // 4M
    const int FIN_BLOCKS = 640;                          // 640*256 thr, multiple of 128

    // ---- degrees and dinv = rsqrt(deg) (deg includes self loop) ----
    fill_kernel<<<cdiv(N_NODES, 256), 256, 0, stream>>>(dinv, 1.0f, N_NODES);
    degree_kernel<<<cdiv(N_EDGES, 256), 256, 0, stream>>>(dst, dinv, N_EDGES);
    rsqrt_inplace_kernel<<<cdiv(N_NODES, 256), 256, 0, stream>>>(dinv, N_NODES);

    const int tilesM = N_NODES / 16;                     // 6250 (exact)

    // ================= layer 1 =================
    {
        constexpr int NGRP = HIDC / 64;                  // NT=4 -> 2 groups
        int waves = tilesM * NGRP;                       // 12500
        pack_w_kernel<HIDC, HIDC><<<cdiv((KDIM/4)*HIDC*4, 256), 256, 0, stream>>>(W1, Wp);
        fill_kernel<<<cdiv(NC, 256), 256, 0, stream>>>(bufB, 0.0f, NC);
        fill_kernel<<<1, 256, 0, stream>>>(stats, 0.0f, 256);
        gemm_wmma_f32<HIDC, HIDC, 4><<<cdiv((long long)waves * 32, 256), 256, 0, stream>>>(
            x, Wp, dinv, bufA, N_NODES);
        scatter_kernel<32><<<cdiv((long long)N_EDGES * 32, 256), 256, 0, stream>>>(
            (const float4*)bufA, src, dst, bufB, N_EDGES);
        finalize_bn_stats_kernel<<<FIN_BLOCKS, 256, 0, stream>>>(
            bufA, bufB, dinv, b1, stats, N_NODES);
        bn_relu_kernel<<<cdiv(NC, 256), 256, 0, stream>>>(bufB, stats, g1, bt1, N_NODES);
    }

    // ================= layer 2 =================
    {
        constexpr int NGRP = HIDC / 64;
        int waves = tilesM * NGRP;
        pack_w_kernel<HIDC, HIDC><<<cdiv((KDIM/4)*HIDC*4, 256), 256, 0, stream>>>(W2, Wp);
        gemm_wmma_f32<HIDC, HIDC, 4><<<cdiv((long long)waves * 32, 256), 256, 0, stream>>>(
            bufB, Wp, dinv, bufA, N_NODES);
        fill_kernel<<<cdiv(NC, 256), 256, 0, stream>>>(bufB, 0.0f, NC);  // after gemm read it
        fill_kernel<<<1, 256, 0, stream>>>(stats, 0.0f, 256);
        scatter_kernel<32><<<cdiv((long long)N_EDGES * 32, 256), 256, 0, stream>>>(
            (const float4*)bufA, src, dst, bufB, N_EDGES);
        finalize_bn_stats_kernel<<<FIN_BLOCKS, 256, 0, stream>>>(
            bufA, bufB, dinv, b2, stats, N_NODES);
        bn_relu_kernel<<<cdiv(NC, 256), 256, 0, stream>>>(bufB, stats, g2, bt2, N_NODES);
    }

    // ================= layer 3 (C_out = 40, padded to 48) =================
    {
        int waves = tilesM;                              // NGRP = 48/(16*3) = 1
        pack_w_kernel<OUTC, 48><<<cdiv((KDIM/4)*48*4, 256), 256, 0, stream>>>(W3, Wp);
        gemm_wmma_f32<OUTC, 48, 3><<<cdiv((long long)waves * 32, 256), 256, 0, stream>>>(
            bufB, Wp, dinv, bufA, N_NODES);
        fill_kernel<<<cdiv(NO, 256), 256, 0, stream>>>(out, 0.0f, NO);
        scatter_kernel<10><<<cdiv((long long)N_EDGES * 10, 256), 256, 0, stream>>>(
            (const float4*)bufA, src, dst, out, N_EDGES);
        finalize_out_kernel<<<cdiv(NO, 256), 256, 0, stream>>>(
            bufA, out, dinv, b3, N_NODES);
    }
}